// HashAutoGenerator_60894046323147
// MI455X (gfx1250) — compile-verified
//
#include <hip/hip_runtime.h>

// ---- problem constants (match reference) ----
#define TSIZE   131072        // hashmap entries per level (2^17)
#define TMASK   (TSIZE - 1)
#define LVLS    16            // levels per group
#define KIN     64            // L*F feature width
#define HIDN    64
#define OUTD    32
#define EGRP    8
#define RRES    32
#define LDP     68            // padded LDS row pitch (floats): 68%64=4 -> conflict-free

typedef __attribute__((ext_vector_type(2))) float v2f;
typedef __attribute__((ext_vector_type(8))) float v8f;

// A-fragment (16x4 f32) from LDS buffer [16][LDP]:
// lanes 0-15: M=lane, {K0,K1}; lanes 16-31: M=lane-16, {K2,K3}
__device__ __forceinline__ v2f load_afrag(const float* buf, int lane, int k0) {
    int m = lane & 15, half = lane >> 4;
    return *(const v2f*)&buf[m * LDP + k0 + 2 * half];
}

// B-fragment (4x16 f32) from row-major global weights w[k][j], leading dim ldw:
// lanes 0-15: N=lane, v0=K0,v1=K1 ; lanes 16-31: N=lane-16, v0=K2,v1=K3
__device__ __forceinline__ v2f load_bfrag(const float* __restrict__ w, int ldw,
                                          int lane, int k0, int j0) {
    int col = lane & 15, half = lane >> 4;
    v2f b;
    b.x = w[(k0 + 2 * half + 0) * ldw + j0 + col];
    b.y = w[(k0 + 2 * half + 1) * ldw + j0 + col];
    return b;
}

__global__ __launch_bounds__(64) void hashgrid_mlp_kernel(
    const float* __restrict__ fc,      // [8, 16, 32, 32]
    const float* __restrict__ tables,  // [8, 16, TSIZE, 4]
    const float* __restrict__ w1,      // [8, 64, 64]
    const float* __restrict__ b1,      // [8, 64]
    const float* __restrict__ w2,      // [8, 64, 64]
    const float* __restrict__ b2,      // [8, 64]
    const float* __restrict__ w3,      // [8, 64, 32]
    const float* __restrict__ b3,      // [8, 32]
    float* __restrict__ out)           // [8192, 256]
{
    __shared__ float bufA[2][16 * LDP];
    __shared__ float bufB[2][16 * LDP];

    const int e      = blockIdx.y;
    const int wave   = threadIdx.x >> 5;
    const int lane   = threadIdx.x & 31;
    const int base_n = blockIdx.x * 32 + wave * 16;   // first sample of this wave's tile
    const int col    = lane & 15;
    const int half   = lane >> 4;

    float* fA = bufA[wave];
    float* fB = bufB[wave];

    // ================= Phase 1: hash-grid features -> fA[16][64] =================
    const float4* tabE = (const float4*)tables + (size_t)e * LVLS * TSIZE;
    #pragma unroll
    for (int p = 0; p < 8; ++p) {
        int task = p * 32 + lane;        // 256 (sample, level) tasks per wave
        int s = task & 15;
        int l = task >> 4;
        int n = base_n + s;
        int x = n & (RRES - 1);
        int y = (n >> 5) & (RRES - 1);
        int b = n >> 10;

        // coords = [fc0, fc1, (x+.5)/R, (y+.5)/R]
        float c0 = fc[((b * 16 + e * 2 + 0) * RRES + y) * RRES + x];
        float c1 = fc[((b * 16 + e * 2 + 1) * RRES + y) * RRES + x];
        float c2 = ((float)x + 0.5f) * (1.0f / RRES);
        float c3 = ((float)y + 0.5f) * (1.0f / RRES);

        // res = floor(16 * 2^(l/15))
        float res = floorf(16.0f * exp2f((float)l * (1.0f / 15.0f)));

        float xs0 = c0 * res, xs1 = c1 * res, xs2 = c2 * res, xs3 = c3 * res;
        float p0 = floorf(xs0), p1 = floorf(xs1), p2 = floorf(xs2), p3 = floorf(xs3);
        unsigned i0 = (unsigned)p0, i1 = (unsigned)p1, i2 = (unsigned)p2, i3 = (unsigned)p3;
        float r0 = xs0 - p0, r1 = xs1 - p1, r2 = xs2 - p2, r3 = xs3 - p3;

        const float4* tab = tabE + (size_t)l * TSIZE;
        float a0 = 0.f, a1 = 0.f, a2 = 0.f, a3 = 0.f;
        #pragma unroll
        for (int c = 0; c < 16; ++c) {
            unsigned o0 = (c >> 3) & 1u, o1 = (c >> 2) & 1u, o2 = (c >> 1) & 1u, o3 = c & 1u;
            unsigned h = ((i0 + o0) * 1u)
                       ^ ((i1 + o1) * 2654435761u)
                       ^ ((i2 + o2) * 805459861u)
                       ^ ((i3 + o3) * 3674653429u);
            h &= TMASK;
            float4 v = tab[h];
            float w = (o0 ? r0 : 1.0f - r0) * (o1 ? r1 : 1.0f - r1)
                    * (o2 ? r2 : 1.0f - r2) * (o3 ? r3 : 1.0f - r3);
            a0 += v.x * w; a1 += v.y * w; a2 += v.z * w; a3 += v.w * w;
        }
        *(float4*)&fA[s * LDP + l * 4] = make_float4(a0, a1, a2, a3);
    }
    __syncthreads();

    // ================= Layer 1: h1 = lrelu(feats @ w1 + b1) -> fB =================
    const float* w1e = w1 + e * (KIN * HIDN);
    const float* b1e = b1 + e * HIDN;
    #pragma unroll
    for (int jt = 0; jt < 4; ++jt) {
        float bias = b1e[jt * 16 + col];
        v8f acc;
        #pragma unroll
        for (int i = 0; i < 8; ++i) acc[i] = bias;
        #pragma unroll
        for (int k = 0; k < 16; ++k) {
            v2f a  = load_afrag(fA, lane, k * 4);
            v2f bb = load_bfrag(w1e, HIDN, lane, k * 4, jt * 16);
            acc = __builtin_amdgcn_wmma_f32_16x16x4_f32(false, a, false, bb,
                                                        (short)0, acc, false, false);
        }
        #pragma unroll
        for (int v = 0; v < 8; ++v) {
            float h = acc[v];
            h = h > 0.0f ? h : 0.2f * h;
            fB[(v + 8 * half) * LDP + jt * 16 + col] = h;
        }
    }
    __syncthreads();

    // ================= Layer 2: h2 = lrelu(h1 @ w2 + b2) -> fA =================
    const float* w2e = w2 + e * (HIDN * HIDN);
    const float* b2e = b2 + e * HIDN;
    #pragma unroll
    for (int jt = 0; jt < 4; ++jt) {
        float bias = b2e[jt * 16 + col];
        v8f acc;
        #pragma unroll
        for (int i = 0; i < 8; ++i) acc[i] = bias;
        #pragma unroll
        for (int k = 0; k < 16; ++k) {
            v2f a  = load_afrag(fB, lane, k * 4);
            v2f bb = load_bfrag(w2e, HIDN, lane, k * 4, jt * 16);
            acc = __builtin_amdgcn_wmma_f32_16x16x4_f32(false, a, false, bb,
                                                        (short)0, acc, false, false);
        }
        #pragma unroll
        for (int v = 0; v < 8; ++v) {
            float h = acc[v];
            h = h > 0.0f ? h : 0.2f * h;
            fA[(v + 8 * half) * LDP + jt * 16 + col] = h;
        }
    }
    __syncthreads();

    // ================= Layer 3: out = h2 @ w3 + b3 -> global =================
    const float* w3e = w3 + e * (HIDN * OUTD);
    const float* b3e = b3 + e * OUTD;
    #pragma unroll
    for (int jt = 0; jt < 2; ++jt) {
        float bias = b3e[jt * 16 + col];
        v8f acc;
        #pragma unroll
        for (int i = 0; i < 8; ++i) acc[i] = bias;
        #pragma unroll
        for (int k = 0; k < 16; ++k) {
            v2f a  = load_afrag(fA, lane, k * 4);
            v2f bb = load_bfrag(w3e, OUTD, lane, k * 4, jt * 16);
            acc = __builtin_amdgcn_wmma_f32_16x16x4_f32(false, a, false, bb,
                                                        (short)0, acc, false, false);
        }
        #pragma unroll
        for (int v = 0; v < 8; ++v) {
            int m = v + 8 * half;   // sample row within tile
            out[(size_t)(base_n + m) * (EGRP * OUTD) + e * OUTD + jt * 16 + col] = acc[v];
        }
    }
}

extern "C" void kernel_launch(void* const* d_in, const int* in_sizes, int n_in,
                              void* d_out, int out_size, void* d_ws, size_t ws_size,
                              hipStream_t stream) {
    const float* fc     = (const float*)d_in[0];
    const float* tables = (const float*)d_in[1];
    const float* w1     = (const float*)d_in[2];
    const float* b1     = (const float*)d_in[3];
    const float* w2     = (const float*)d_in[4];
    const float* b2     = (const float*)d_in[5];
    const float* w3     = (const float*)d_in[6];
    const float* b3     = (const float*)d_in[7];
    float* out = (float*)d_out;

    // N = 8*32*32 = 8192 samples; 32 samples per block (2 waves x 16), E groups in grid.y
    dim3 grid(8192 / 32, EGRP);
    dim3 block(64);
    hashgrid_mlp_kernel<<<grid, block, 0, stream>>>(fc, tables, w1, b1, w2, b2, w3, b3, out);
}